// StrongAugment_17746804867715
// MI455X (gfx1250) — compile-verified
//
#include <hip/hip_runtime.h>
#include <hip/hip_bf16.h>

typedef __attribute__((ext_vector_type(2))) float    v2f;
typedef __attribute__((ext_vector_type(8))) float    v8f;
typedef __attribute__((ext_vector_type(4))) unsigned v4u;
typedef __attribute__((ext_vector_type(4))) int      v4i;
typedef __attribute__((ext_vector_type(8))) int      v8i;

#define HH 32
#define WW 32
#define CROPN 25
#define PITCH 33                          // padded LDS row (matches TDM pad: 32dw + 1dw)
#define IDX(c,y,x) ((c)*(HH*PITCH) + (y)*PITCH + (x))

// therock-10.0 headers (amdgpu-toolchain) ship this header and the 6-arg TDM builtins;
// ROCm 7.2 lacks the header and uses the 5-arg builtins.
#if defined(__has_include)
#if __has_include(<hip/amd_detail/amd_gfx1250_TDM.h>)
#define TDM_SIX_ARGS 1
#else
#define TDM_SIX_ARGS 0
#endif
#else
#define TDM_SIX_ARGS 0
#endif

#if __has_builtin(__builtin_amdgcn_tensor_load_to_lds)
#define HAVE_TDM_LD 1
#else
#define HAVE_TDM_LD 0
#endif
#if __has_builtin(__builtin_amdgcn_tensor_store_from_lds)
#define HAVE_TDM_ST 1
#else
#define HAVE_TDM_ST 0
#endif

__constant__ float cMEAN[3] = {0.5071f, 0.4867f, 0.4408f};
__constant__ float cSTD[3]  = {0.2675f, 0.2565f, 0.2761f};

__device__ __forceinline__ unsigned hash_u32(unsigned x) {
    x ^= x >> 16; x *= 0x7feb352dU;
    x ^= x >> 15; x *= 0x846ca68bU;
    x ^= x >> 16; return x;
}
__device__ __forceinline__ float rng01(unsigned img, unsigned salt) {
    return (float)(hash_u32(img * 0x9E3779B9U + salt * 0x85EBCA77U + 0x42U) >> 8)
           * (1.0f / 16777216.0f);
}

__device__ __forceinline__ float reflectf(float c, float size) {
    const float lo = -0.5f;
    float t     = fabsf(c - lo);
    float extra = fmodf(t, size);
    float flips = floorf(t / size);
    float out   = (fmodf(flips, 2.0f) == 0.0f) ? (lo + extra) : (lo + size - extra);
    return fminf(fmaxf(out, 0.0f), size - 1.0f);
}

// 32x32 separable box-blur matrix (edge-padded 1/3 band): V[0][0]=V[31][31]=2/3
__device__ __forceinline__ float vmat(int i, int j) {
    float v = ((i - j <= 1) && (j - i <= 1)) ? (1.0f / 3.0f) : 0.0f;
    if ((i == 0 && j == 0) || (i == 31 && j == 31)) v += 1.0f / 3.0f;
    return v;
}

// Build a simple dense-1D (optionally LDS-padded) TDM descriptor group pair.
__device__ __forceinline__ void make_tdm_desc(v4u& g0, v8i& g1,
                                              unsigned lds_addr,
                                              unsigned long long gaddr,
                                              int pad /*1: +1dw per 32dw on LDS side*/) {
    g0.x = 1u;                                   // count=1 (valid user descriptor)
    g0.y = lds_addr;                             // lds_addr (bytes)
    g0.z = (unsigned)(gaddr & 0xFFFFFFFFu);      // global_addr[31:0]
    g0.w = (unsigned)((gaddr >> 32) & 0x1FFFFFFu)// global_addr[56:32]
         | 0x80000000u;                          // type=2 ("image") in bits[127:126]
    unsigned w0 = (2u << 16);                    // data_size = 4 bytes
    if (pad) w0 |= (1u << 20) | (4u << 22);      // pad_enable, interval=32dw, amount=1dw
    g1[0] = (int)w0;
    g1[1] = (int)(3072u << 16);                  // tensor_dim0[15:0] (=3072)
    g1[2] = (int)(1u << 16);                     // tensor_dim0 hi=0 | tensor_dim1=1
    g1[3] = (int)(3072u << 16);                  // tensor_dim1 hi=0 | tile_dim0=3072
    g1[4] = 1;                                   // tile_dim1=1 | tile_dim2=0
    g1[5] = 3072;                                // tensor_dim0_stride[31:0]
    g1[6] = 0;
    g1[7] = 0;
}

__global__ __launch_bounds__(256)
void augment_kernel(const float* __restrict__ xin,
                    const float* __restrict__ aff,
                    const int*   __restrict__ applyCrop,
                    float* __restrict__ outImg,
                    float* __restrict__ outFlip,
                    float* __restrict__ outAff) {
    __shared__ float bufA[3 * HH * PITCH];
    __shared__ float bufB[3 * HH * PITCH];
    __shared__ float vms[HH * PITCH];           // blur matrix V, LDS-resident
#if HAVE_TDM_ST
    __shared__ float outb[3072];                // dense staging for TDM store
#endif

    const int img = blockIdx.x;
    const int t   = threadIdx.x;
    const int wave = t >> 5;
    const unsigned u = (unsigned)img;

    // ---- stage 0a: TDM async DMA of the raw image HBM -> LDS (wave 0 issues) ----
    // Pad feature inserts 1 DWORD after every 32 DWORDs => pitch-33 rows for free.
#if HAVE_TDM_LD
    if (wave == 0) {
        v4u g0; v8i g1;
        make_tdm_desc(g0, g1, (unsigned)(uintptr_t)(&bufA[0]),
                      (unsigned long long)(uintptr_t)(xin + (size_t)img * 3072), 1);
        v4i gz = {0, 0, 0, 0};
#if TDM_SIX_ARGS
        v8i gz8 = {0, 0, 0, 0, 0, 0, 0, 0};
        __builtin_amdgcn_tensor_load_to_lds(g0, g1, gz, gz, gz8, 0);
#else
        __builtin_amdgcn_tensor_load_to_lds(g0, g1, gz, gz, 0);
#endif
    }
#else
    {
        const float* gsrc = xin + (size_t)img * 3072;
#pragma unroll
        for (int i = 0; i < 12; ++i) {
            int idx = t + i * 256;
            int c = idx >> 10, rem = idx & 1023;
            bufA[IDX(c, rem >> 5, rem & 31)] = gsrc[idx];
        }
    }
#endif

    // ---- stage 0b (overlaps the DMA): build V in LDS; per-image params ----
#pragma unroll
    for (int i = 0; i < 4; ++i) {
        int idx = t + i * 256;
        vms[(idx >> 5) * PITCH + (idx & 31)] = vmat(idx >> 5, idx & 31);
    }

    const int   top  = (int)(hash_u32(u * 2654435761u + 101u) & 7u);
    const int   left = (int)(hash_u32(u * 2654435761u + 202u) & 7u);
    const int   flip = rng01(u, 3) < 0.5f;
    const float br   = 1.0f + (rng01(u, 4) - 0.5f) * 0.8f;
    const float ct   = 1.0f + (rng01(u, 5) - 0.5f) * 0.8f;
    const float st   = 1.0f + (rng01(u, 6) - 0.5f) * 0.8f;
    const int   hue  = rng01(u, 7) < 0.5f;
    const int   gry  = rng01(u, 8) < 0.2f;
    const int   blr  = rng01(u, 9) < 0.5f;
    const int   sol  = rng01(u, 10) < 0.1f;
    const int   docrop = (applyCrop[0] != 0);

    const float th0 = aff[(size_t)img * 6 + 0], th1 = aff[(size_t)img * 6 + 1];
    const float th2 = aff[(size_t)img * 6 + 2], th3 = aff[(size_t)img * 6 + 3];
    const float th4 = aff[(size_t)img * 6 + 4], th5 = aff[(size_t)img * 6 + 5];

#if HAVE_TDM_LD
    if (wave == 0) __builtin_amdgcn_s_wait_tensorcnt(0);
#endif
    __syncthreads();

    // NOTE: denormalize (x*STD+MEAN) commutes through bilinear/affine sampling
    // (convex weights), so stages 1-2 operate on RAW values; denorm folds into stage 3.

    // ---- stage 1: crop(25x25)+bilinear resize back to 32x32, A -> B ----
    if (docrop) {
#pragma unroll
        for (int i = 0; i < 4; ++i) {
            int p = t + i * 256;
            int y = p >> 5, x = p & 31;
            float dy = fmaxf((y + 0.5f) * (25.0f / 32.0f) - 0.5f, 0.0f);
            float dx = fmaxf((x + 0.5f) * (25.0f / 32.0f) - 0.5f, 0.0f);
            int i0y = (int)dy; float ly = dy - (float)i0y; int i1y = min(i0y + 1, CROPN - 1);
            int i0x = (int)dx; float lx = dx - (float)i0x; int i1x = min(i0x + 1, CROPN - 1);
            int r0 = top + i0y, r1 = top + i1y, c0 = left + i0x, c1 = left + i1x;
#pragma unroll
            for (int c = 0; c < 3; ++c) {
                float v00 = bufA[IDX(c, r0, c0)], v01 = bufA[IDX(c, r0, c1)];
                float v10 = bufA[IDX(c, r1, c0)], v11 = bufA[IDX(c, r1, c1)];
                bufB[IDX(c, y, x)] = v00 * (1 - ly) * (1 - lx) + v01 * (1 - ly) * lx
                                   + v10 * ly * (1 - lx)       + v11 * ly * lx;
            }
        }
    } else {
#pragma unroll
        for (int i = 0; i < 12; ++i) {
            int idx = t + i * 256;
            int c = idx >> 10, rem = idx & 1023;
            bufB[IDX(c, rem >> 5, rem & 31)] = bufA[IDX(c, rem >> 5, rem & 31)];
        }
    }
    __syncthreads();

    // ---- stage 2: hflip fused into affine grid-sample (reflect pad), B -> A ----
#pragma unroll
    for (int i = 0; i < 4; ++i) {
        int p = t + i * 256;
        int y = p >> 5, x = p & 31;
        float xn = 2.0f * (x + 0.5f) / 32.0f - 1.0f;
        float yn = 2.0f * (y + 0.5f) / 32.0f - 1.0f;
        float gx = th0 * xn + th1 * yn + th2;
        float gy = th3 * xn + th4 * yn + th5;
        float ix = reflectf(((gx + 1.0f) * 32.0f - 1.0f) * 0.5f, 32.0f);
        float iy = reflectf(((gy + 1.0f) * 32.0f - 1.0f) * 0.5f, 32.0f);
        float fy = floorf(iy), fx = floorf(ix);
        float wy = iy - fy, wx = ix - fx;
        int y0i = min(max((int)fy, 0), 31); int y1i = min(y0i + 1, 31);
        int x0i = min(max((int)fx, 0), 31); int x1i = min(x0i + 1, 31);
        int sx0 = flip ? (31 - x0i) : x0i;       // sampling the flipped image
        int sx1 = flip ? (31 - x1i) : x1i;
#pragma unroll
        for (int c = 0; c < 3; ++c) {
            float v00 = bufB[IDX(c, y0i, sx0)], v01 = bufB[IDX(c, y0i, sx1)];
            float v10 = bufB[IDX(c, y1i, sx0)], v11 = bufB[IDX(c, y1i, sx1)];
            bufA[IDX(c, y, x)] = v00 * (1 - wy) * (1 - wx) + v01 * (1 - wy) * wx
                               + v10 * wy * (1 - wx)       + v11 * wy * wx;
        }
    }
    __syncthreads();

    // ---- stage 3: denorm + brightness/contrast/saturation/hue/gray, A -> B ----
#pragma unroll
    for (int i = 0; i < 4; ++i) {
        int p = t + i * 256;
        int y = p >> 5, x = p & 31;
        float r = bufA[IDX(0, y, x)] * cSTD[0] + cMEAN[0];
        float g = bufA[IDX(1, y, x)] * cSTD[1] + cMEAN[1];
        float b = bufA[IDX(2, y, x)] * cSTD[2] + cMEAN[2];
        r *= br; g *= br; b *= br;
        float gr = (r + g + b) * (1.0f / 3.0f);
        r = gr + (r - gr) * ct; g = gr + (g - gr) * ct; b = gr + (b - gr) * ct;
        gr = (r + g + b) * (1.0f / 3.0f);
        r = gr + (r - gr) * st; g = gr + (g - gr) * st; b = gr + (b - gr) * st;
        if (hue) { float tmp = r; r = b; b = tmp; }     // channel reversal
        if (gry) { gr = (r + g + b) * (1.0f / 3.0f); r = g = b = gr; }
        bufB[IDX(0, y, x)] = r; bufB[IDX(1, y, x)] = g; bufB[IDX(2, y, x)] = b;
    }
    __syncthreads();

    // ---- stage 4: 3x3 box blur as Y = V * X * V^T on the matrix pipe ----
    // Each wave owns one 16x16 tile (same tile for both of its channel jobs),
    // so the V-side WMMA operands are loaded from LDS ONCE per wave.
    const int lane  = t & 31;
    const int lhalf = lane >> 4;     // A: K half; C/D: +8 row half
    const int l15   = lane & 15;
    const int tile  = wave & 3;
    const int mi = (tile >> 1) * 16, ni = (tile & 1) * 16;

    // 4a operand: A = V rows of this tile (per ISA 16x4 f32 A layout)
    v2f va[8];
#pragma unroll
    for (int kc = 0; kc < 8; ++kc) {
        int kh = kc * 4 + lhalf * 2;
        va[kc].x = vms[(mi + l15) * PITCH + kh];
        va[kc].y = vms[(mi + l15) * PITCH + kh + 1];
    }

    // 4a: T = V * X ; X in bufB, T -> bufA
#pragma unroll
    for (int jp = 0; jp < 2; ++jp) {
        int job = wave + jp * 8;               // wave-uniform: EXEC stays full
        if (job < 12) {
            int c = job >> 2;
            v8f acc = {};
#pragma unroll
            for (int kc = 0; kc < 8; ++kc) {
                int kh = kc * 4 + lhalf * 2;
                v2f b;
                b.x = bufB[IDX(c, kh,     ni + l15)];   // B 4x16 layout
                b.y = bufB[IDX(c, kh + 1, ni + l15)];
                acc = __builtin_amdgcn_wmma_f32_16x16x4_f32(
                        false, va[kc], false, b, (short)0, acc, false, false);
            }
#pragma unroll
            for (int v = 0; v < 8; ++v)
                bufA[IDX(c, mi + v + lhalf * 8, ni + l15)] = acc[v];
        }
    }
    __syncthreads();

    // 4b operand: B = (V^T) chunks: b[k][n] = V[ni+l15][k]
    v2f vb[8];
#pragma unroll
    for (int kc = 0; kc < 8; ++kc) {
        int kh = kc * 4 + lhalf * 2;
        vb[kc].x = vms[(ni + l15) * PITCH + kh];
        vb[kc].y = vms[(ni + l15) * PITCH + kh + 1];
    }

    // 4b: Y = T * V^T ; fuse blur-select, solarize, clip, renorm; stage result
#pragma unroll
    for (int jp = 0; jp < 2; ++jp) {
        int job = wave + jp * 8;
        if (job < 12) {
            int c = job >> 2;
            v8f acc = {};
#pragma unroll
            for (int kc = 0; kc < 8; ++kc) {
                int kh = kc * 4 + lhalf * 2;
                v2f a;
                a.x = bufA[IDX(c, mi + l15, kh)];        // T rows (A 16x4 layout)
                a.y = bufA[IDX(c, mi + l15, kh + 1)];
                acc = __builtin_amdgcn_wmma_f32_16x16x4_f32(
                        false, a, false, vb[kc], (short)0, acc, false, false);
            }
            const float mn = cMEAN[c], sd = cSTD[c];
#pragma unroll
            for (int v = 0; v < 8; ++v) {
                int row = mi + v + lhalf * 8;
                int col = ni + l15;
                float xv = blr ? acc[v] : bufB[IDX(c, row, col)];
                if (sol && xv > 0.5f) xv = 1.0f - xv;
                xv = fminf(fmaxf(xv, 0.0f), 1.0f);
                xv = (xv - mn) / sd;
#if HAVE_TDM_ST
                outb[c * 1024 + row * 32 + col] = xv;    // dense: 16-dword runs/half-wave
#else
                outImg[(size_t)img * 3072 + c * 1024 + row * 32 + col] = xv;
#endif
            }
        }
    }

#if HAVE_TDM_ST
    // ---- stage 5: one linear TDM DMA store of the finished image LDS -> HBM ----
    __syncthreads();
    if (wave == 0) {
        v4u g0; v8i g1;
        make_tdm_desc(g0, g1, (unsigned)(uintptr_t)(&outb[0]),
                      (unsigned long long)(uintptr_t)(outImg + (size_t)img * 3072), 0);
        v4i gz = {0, 0, 0, 0};
#if TDM_SIX_ARGS
        v8i gz8 = {0, 0, 0, 0, 0, 0, 0, 0};
        __builtin_amdgcn_tensor_store_from_lds(g0, g1, gz, gz, gz8, 0);
#else
        __builtin_amdgcn_tensor_store_from_lds(g0, g1, gz, gz, 0);
#endif
        // no explicit s_wait_tensorcnt: S_ENDPGM's implicit wait-idle drains TENSORcnt
    }
#endif

    // ---- auxiliary outputs: flip_mask + shared_affine passthrough ----
    if (t == 0) outFlip[img] = flip ? 1.0f : 0.0f;
    if (t < 6)  outAff[(size_t)img * 6 + t] = aff[(size_t)img * 6 + t];
}

extern "C" void kernel_launch(void* const* d_in, const int* in_sizes, int n_in,
                              void* d_out, int out_size, void* d_ws, size_t ws_size,
                              hipStream_t stream) {
    (void)n_in; (void)out_size; (void)d_ws; (void)ws_size;
    const float* x      = (const float*)d_in[0];
    const float* aff    = (const float*)d_in[1];
    const int*   apcrop = (const int*)d_in[2];
    const int    Btot   = in_sizes[0] / 3072;
    float* out     = (float*)d_out;
    float* outFlip = out + (size_t)Btot * 3072;
    float* outAff  = outFlip + Btot;
    augment_kernel<<<Btot, 256, 0, stream>>>(x, aff, apcrop, out, outFlip, outAff);
}